// Dipole_2078764171303
// MI455X (gfx1250) — compile-verified
//
#include <hip/hip_runtime.h>
#include <math.h>

#define BB   32
#define TT   100
#define VV   10000
#define DD   128
#define OUTN 1000
#define G3   (3*DD)   // 384
#define WPAD 132      // padded LDS row (floats) to avoid 16-way bank conflicts

typedef __attribute__((ext_vector_type(2))) float v2f;
typedef __attribute__((ext_vector_type(8))) float v8f;

__device__ __forceinline__ v8f wmma4(v2f a, v2f b, v8f c) {
  // V_WMMA_F32_16X16X4_F32 : D = A(16x4 f32) x B(4x16 f32) + C(16x16 f32)
  return __builtin_amdgcn_wmma_f32_16x16x4_f32(false, a, false, b, (short)0, c, false, false);
}

__device__ __forceinline__ float sigmoidf_(float x) { return 1.0f / (1.0f + expf(-x)); }

// ---------------------------------------------------------------------------
// Kernel A: sparse multi-hot embedding  e[bt,d] = sum_v x[bt,v]*emb[v,d]
// one block per (b,t); compact nonzero codes into LDS, then coalesced row sum.
// ---------------------------------------------------------------------------
__global__ void embed_kernel(const float* __restrict__ x, const float* __restrict__ emb,
                             float* __restrict__ e, int* __restrict__ mask)
{
  __shared__ int   s_idx[1024];
  __shared__ float s_val[1024];
  __shared__ int   s_cnt;
  __shared__ int   s_flag;
  const int bt  = blockIdx.x;
  const int tid = threadIdx.x;        // blockDim.x == 128
  if (tid == 0) { s_cnt = 0; s_flag = 0; }
  __syncthreads();
  const float* row = x + (size_t)bt * VV;
  for (int v = tid; v < VV; v += 128) {
    float xv = row[v];
    if (xv != 0.0f) {
      int p = atomicAdd(&s_cnt, 1);
      if (p < 1024) { s_idx[p] = v; s_val[p] = xv; }
    }
  }
  __syncthreads();
  int n = s_cnt; if (n > 1024) n = 1024;
  float acc = 0.0f;
  const int d = tid;
  for (int i = 0; i < n; ++i)
    acc += s_val[i] * emb[(size_t)s_idx[i] * DD + d];
  e[(size_t)bt * DD + d] = acc;
  if (acc != 0.0f) s_flag = 1;
  __syncthreads();
  if (tid == 0) mask[bt] = s_flag;
}

// ---------------------------------------------------------------------------
// Kernel B: C[M,384] = A[M,128] @ W[384,128]^T + bias   (M = B*T = 3200)
// one wave per (16 x 64) strip: 4 N-tiles share each A fragment (A reuse x4).
// ---------------------------------------------------------------------------
__global__ void xgemm_kernel(const float* __restrict__ A, const float* __restrict__ W,
                             const float* __restrict__ bias, float* __restrict__ C)
{
  const int mt = blockIdx.x;            // 0..199
  const int n0 = blockIdx.y * 4;        // N-tiles n0..n0+3 (blockIdx.y 0..5)
  const int lane = threadIdx.x;         // blockDim.x == 32
  const int lh = lane >> 4, lm = lane & 15;
  const float* arow = A + (size_t)(mt * 16 + lm) * DD;        // A[m][k]
  const float* brow[4];
#pragma unroll
  for (int j = 0; j < 4; ++j)
    brow[j] = W + (size_t)((n0 + j) * 16 + lm) * DD;          // W[n][k] = B^T
  v8f c[4] = {{}, {}, {}, {}};
  for (int kk = 0; kk < DD; kk += 4) {
    const int k = kk + 2 * lh;          // ISA A/B layout: k = kbase + 2*(lane/16) + v
    v2f a; a.x = arow[k]; a.y = arow[k + 1];
#pragma unroll
    for (int j = 0; j < 4; ++j) {
      v2f b; b.x = brow[j][k]; b.y = brow[j][k + 1];
      c[j] = wmma4(a, b, c[j]);
    }
  }
#pragma unroll
  for (int j = 0; j < 4; ++j) {
    const int col = (n0 + j) * 16 + lm;
    const float bv = bias[col];
    for (int i = 0; i < 8; ++i) {
      const int rowi = mt * 16 + i + 8 * lh;   // C layout: VGPR i -> rows i / i+8
      C[(size_t)rowi * G3 + col] = c[j][i] + bv;
    }
  }
}

// ---------------------------------------------------------------------------
// Kernel C: persistent GRU scan, one block per direction (256 thr = 8 waves).
// whh (192 KB) staged once into CDNA5's 320 KB LDS (padded rows -> no bank
// conflicts); h lives in LDS; per step each wave does 6 WMMA tiles
// (2 m-tiles x 3 gates) for its 16-wide d-slice, then the full r/z/n/h
// update happens lane-locally. global_prefetch for next step's x gates.
// ---------------------------------------------------------------------------
__global__ void gru_kernel(const float* __restrict__ x3f, const float* __restrict__ x3b,
                           const float* __restrict__ whh_f, const float* __restrict__ bhh_f,
                           const float* __restrict__ whh_b, const float* __restrict__ bhh_b,
                           float* __restrict__ hf, float* __restrict__ hb)
{
  __shared__ float w_lds[G3 * WPAD];   // 384*132*4 = 202,752 B
  __shared__ float h_lds[BB * WPAD];   //  32*132*4 =  16,896 B   (total ~214.5 KB < 320 KB)
  const int dir = blockIdx.x;
  const float* x3  = dir ? x3b   : x3f;
  const float* whh = dir ? whh_b : whh_f;
  const float* bhh = dir ? bhh_b : bhh_f;
  float* hout      = dir ? hb    : hf;

  const int tid  = threadIdx.x;
  const int w    = tid >> 5;          // wave = d-tile 0..7
  const int lane = tid & 31;
  const int lh   = lane >> 4, lm = lane & 15;

  // stage whh into LDS (row-major with padded stride), zero h
  for (int i = tid; i < G3 * DD; i += 256) {
    const int r = i >> 7, col = i & (DD - 1);   // i/128, i%128
    w_lds[r * WPAD + col] = whh[i];
  }
  for (int i = tid; i < BB * WPAD; i += 256) h_lds[i] = 0.0f;
  __syncthreads();

  const int dcol = w * 16 + lm;
  const float b0 = bhh[dcol], b1 = bhh[DD + dcol], b2 = bhh[2 * DD + dcol];
  const float* wr0 = w_lds + (size_t)(0 * DD + dcol) * WPAD;  // whh[g*128+dcol][k]
  const float* wr1 = w_lds + (size_t)(1 * DD + dcol) * WPAD;
  const float* wr2 = w_lds + (size_t)(2 * DD + dcol) * WPAD;

  for (int step = 0; step < TT; ++step) {
    const int t = dir ? (TT - 1 - step) : step;
    // prefetch next step's x gates into cache (global_prefetch_b8)
    if (step + 1 < TT) {
      const int tn = dir ? (TT - 2 - step) : (step + 1);
      const int mpf = (tid >> 3) & 31;                 // cover all 32 batch rows
      __builtin_prefetch(&x3[((size_t)mpf * TT + tn) * G3 + dcol], 0, 1);
    }
    v8f c00 = {}, c01 = {}, c02 = {}, c10 = {}, c11 = {}, c12 = {};
    for (int kk = 0; kk < DD; kk += 4) {
      const int k = kk + 2 * lh;
      v2f a0, a1, bf;
      a0.x = h_lds[lm * WPAD + k];        a0.y = h_lds[lm * WPAD + k + 1];
      a1.x = h_lds[(16 + lm) * WPAD + k]; a1.y = h_lds[(16 + lm) * WPAD + k + 1];
      bf.x = wr0[k]; bf.y = wr0[k + 1];
      c00 = wmma4(a0, bf, c00); c10 = wmma4(a1, bf, c10);
      bf.x = wr1[k]; bf.y = wr1[k + 1];
      c01 = wmma4(a0, bf, c01); c11 = wmma4(a1, bf, c11);
      bf.x = wr2[k]; bf.y = wr2[k + 1];
      c02 = wmma4(a0, bf, c02); c12 = wmma4(a1, bf, c12);
    }
    __syncthreads();                  // all h_lds reads of this step done

    float hnew[2][8];
    for (int mt = 0; mt < 2; ++mt) {
      const v8f cr = mt ? c10 : c00;
      const v8f cz = mt ? c11 : c01;
      const v8f cn = mt ? c12 : c02;
      for (int i = 0; i < 8; ++i) {
        const int m = mt * 16 + i + 8 * lh;          // batch index
        const size_t xrow = ((size_t)m * TT + t) * G3;
        const float xr = x3[xrow + dcol];
        const float xz = x3[xrow + DD + dcol];
        const float xn = x3[xrow + 2 * DD + dcol];
        const float r  = sigmoidf_(xr + cr[i] + b0);
        const float z  = sigmoidf_(xz + cz[i] + b1);
        const float nn = tanhf(xn + r * (cn[i] + b2));
        const float hp = h_lds[m * WPAD + dcol];     // owned by this lane only
        hnew[mt][i] = (1.0f - z) * nn + z * hp;
      }
    }
    for (int mt = 0; mt < 2; ++mt)
      for (int i = 0; i < 8; ++i) {
        const int m = mt * 16 + i + 8 * lh;
        h_lds[m * WPAD + dcol] = hnew[mt][i];
        hout[((size_t)m * TT + t) * DD + dcol] = hnew[mt][i];
      }
    __syncthreads();                  // h_lds update visible before next step
  }
}

// ---------------------------------------------------------------------------
// Kernel D: attention + combine. one block per batch row.
// ---------------------------------------------------------------------------
__global__ void attn_kernel(const float* __restrict__ hf, const float* __restrict__ hb,
                            const int* __restrict__ mask,
                            const float* __restrict__ attn_w, const float* __restrict__ attn_b,
                            const float* __restrict__ comb_w, const float* __restrict__ comb_b,
                            float* __restrict__ feat)
{
  __shared__ float s_sc[TT];
  __shared__ float s_a[TT];
  __shared__ float s_cat[4 * DD];
  __shared__ int   s_last;
  const int b = blockIdx.x, tid = threadIdx.x;   // blockDim.x == 256

  for (int t = tid; t < TT; t += 256) {
    float s = attn_b[0];
    const size_t base = ((size_t)b * TT + t) * DD;
    for (int d = 0; d < DD; ++d)
      s += hf[base + d] * attn_w[d] + hb[base + d] * attn_w[DD + d];
    if (mask[b * TT + t] == 0) s = -1e9f;
    s_sc[t] = s;
  }
  __syncthreads();
  if (tid == 0) {
    float mx = s_sc[0]; int cnt = 0;
    for (int t = 0; t < TT; ++t) { if (s_sc[t] > mx) mx = s_sc[t]; cnt += mask[b * TT + t]; }
    float ssum = 0.0f;
    for (int t = 0; t < TT; ++t) { float ev = expf(s_sc[t] - mx); s_a[t] = ev; ssum += ev; }
    const float inv = 1.0f / ssum;
    for (int t = 0; t < TT; ++t) s_a[t] *= inv;
    s_last = cnt - 1;
  }
  __syncthreads();
  const int last = s_last;
  if (tid < 2 * DD) {
    const int f = tid;
    const float* hsrc = (f < DD) ? hf : hb;
    const int d = (f < DD) ? f : f - DD;
    float cf = 0.0f;
    for (int t = 0; t < TT; ++t)
      cf += s_a[t] * hsrc[((size_t)b * TT + t) * DD + d];
    s_cat[f] = cf;                                              // context
    s_cat[2 * DD + f] = hsrc[((size_t)b * TT + last) * DD + d]; // h_last
  }
  __syncthreads();
  if (tid < DD) {
    float acc = comb_b[tid];
    for (int f = 0; f < 4 * DD; ++f)
      acc += s_cat[f] * comb_w[(size_t)tid * (4 * DD) + f];
    feat[(size_t)b * DD + tid] = tanhf(acc);
  }
}

// ---------------------------------------------------------------------------
// Kernel E: final FC  logits[b,o] = feat[b,:] . fc_w[o,:] + fc_b[o]
// ---------------------------------------------------------------------------
__global__ void fc_kernel(const float* __restrict__ feat, const float* __restrict__ fc_w,
                          const float* __restrict__ fc_b, float* __restrict__ out)
{
  const int idx = blockIdx.x * blockDim.x + threadIdx.x;
  if (idx >= BB * OUTN) return;
  const int b = idx / OUTN, o = idx % OUTN;
  float acc = fc_b[o];
  const float* fr = feat + (size_t)b * DD;
  const float* wr = fc_w + (size_t)o * DD;
  for (int d = 0; d < DD; ++d) acc += fr[d] * wr[d];
  out[idx] = acc;
}

// ---------------------------------------------------------------------------
extern "C" void kernel_launch(void* const* d_in, const int* in_sizes, int n_in,
                              void* d_out, int out_size, void* d_ws, size_t ws_size,
                              hipStream_t stream) {
  (void)in_sizes; (void)n_in; (void)out_size; (void)ws_size;
  const float* batchdata = (const float*)d_in[0];
  const float* emb    = (const float*)d_in[1];
  const float* wih_f  = (const float*)d_in[2];
  const float* whh_f  = (const float*)d_in[3];
  const float* bih_f  = (const float*)d_in[4];
  const float* bhh_f  = (const float*)d_in[5];
  const float* wih_b  = (const float*)d_in[6];
  const float* whh_b  = (const float*)d_in[7];
  const float* bih_b  = (const float*)d_in[8];
  const float* bhh_b  = (const float*)d_in[9];
  const float* attn_w = (const float*)d_in[10];
  const float* attn_b = (const float*)d_in[11];
  const float* comb_w = (const float*)d_in[12];
  const float* comb_b = (const float*)d_in[13];
  const float* fc_w   = (const float*)d_in[14];
  const float* fc_b   = (const float*)d_in[15];
  float* out = (float*)d_out;

  // workspace layout (floats)
  float* ws   = (float*)d_ws;
  float* e    = ws;                         // 3200*128
  float* xf   = ws + 409600;                // 3200*384
  float* xb   = ws + 1638400;               // 3200*384
  float* hf   = ws + 2867200;               // 3200*128
  float* hb   = ws + 3276800;               // 3200*128
  float* feat = ws + 3686400;               // 32*128
  int*   mask = (int*)(ws + 3690496);       // 3200

  embed_kernel<<<BB * TT, 128, 0, stream>>>(batchdata, emb, e, mask);

  dim3 gtiles((BB * TT) / 16, G3 / (16 * 4));   // 200 x 6, 4 N-tiles per wave
  xgemm_kernel<<<gtiles, 32, 0, stream>>>(e, wih_f, bih_f, xf);
  xgemm_kernel<<<gtiles, 32, 0, stream>>>(e, wih_b, bih_b, xb);

  gru_kernel<<<2, 256, 0, stream>>>(xf, xb, whh_f, bhh_f, whh_b, bhh_b, hf, hb);

  attn_kernel<<<BB, 256, 0, stream>>>(hf, hb, mask, attn_w, attn_b, comb_w, comb_b, feat);

  fc_kernel<<<(BB * OUTN + 255) / 256, 256, 0, stream>>>(feat, fc_w, fc_b, out);
}